// SP_GNNStage_71863392796753
// MI455X (gfx1250) — compile-verified
//
#include <hip/hip_runtime.h>
#include <math.h>

#define N_NODES 50000
#define N_EDGES 640000
#define DIM     128
#define N_LAYERS 4
#define K_MAX    4

typedef float v2f __attribute__((ext_vector_type(2)));
typedef float v8f __attribute__((ext_vector_type(8)));

// ---------------- one-time setup kernels ----------------

__global__ void k_copy(const float* __restrict__ in, float* __restrict__ out, int n) {
  int i = blockIdx.x * blockDim.x + threadIdx.x;
  if (i < n) out[i] = in[i];
}

__global__ void k_fill1(float* __restrict__ p, int n) {
  int i = blockIdx.x * blockDim.x + threadIdx.x;
  if (i < n) p[i] = 1.0f;
}

__global__ void k_deg_accum(const int* __restrict__ dst, const int* __restrict__ ea,
                            float* __restrict__ deg, int E) {
  int i = blockIdx.x * blockDim.x + threadIdx.x;
  if (i < E) {
    int k = ea[i] - 1;
    unsafeAtomicAdd(&deg[dst[i] * K_MAX + k], 1.0f);
  }
}

__global__ void k_rsqrt_inplace(float* __restrict__ p, int n) {
  int i = blockIdx.x * blockDim.x + threadIdx.x;
  if (i < n) p[i] = rsqrtf(p[i]);
}

__global__ void k_edge_norm(const int* __restrict__ src, const int* __restrict__ dst,
                            const int* __restrict__ ea, const float* __restrict__ dinv,
                            float* __restrict__ norm_e, int E) {
  int i = blockIdx.x * blockDim.x + threadIdx.x;
  if (i < E) {
    int k = ea[i] - 1;
    norm_e[i] = dinv[src[i] * K_MAX + k] * dinv[dst[i] * K_MAX + k];
  }
}

// softmax over K=4 for all layers; trivial scalar work
__global__ void k_softmax(const float* __restrict__ alpha, float* __restrict__ a_all) {
  if (blockIdx.x == 0 && threadIdx.x == 0) {
    for (int t = 0; t < N_LAYERS; ++t) {
      float m = alpha[t * K_MAX];
      for (int k = 1; k < K_MAX; ++k) m = fmaxf(m, alpha[t * K_MAX + k]);
      float e[K_MAX], s = 0.0f;
      for (int k = 0; k < K_MAX; ++k) { e[k] = __expf(alpha[t * K_MAX + k] - m); s += e[k]; }
      float inv = 1.0f / s;
      for (int k = 0; k < K_MAX; ++k) a_all[t * K_MAX + k] = e[k] * inv;
    }
  }
}

// self_coef[n] = sum_k a[t,k] / deg[n,k] = sum_k a[t,k] * dinv[n,k]^2
__global__ void k_selfcoef(const float* __restrict__ dinv, const float* __restrict__ a_all,
                           float* __restrict__ sc, int t) {
  int n = blockIdx.x * blockDim.x + threadIdx.x;
  if (n < N_NODES) {
    float s = 0.0f;
#pragma unroll
    for (int k = 0; k < K_MAX; ++k) {
      float di = dinv[n * K_MAX + k];
      s = fmaf(a_all[t * K_MAX + k], di * di, s);
    }
    sc[n] = s;
  }
}

// ---------------- per-layer kernels ----------------

// H[M,128] = X[M,128] @ W[128,128]  via V_WMMA_F32_16X16X4_F32.
// One wave per 16x16 tile; 8 waves/block = one 16-row block across all 8 N-tiles.
// A 16x4 f32 layout: lanes 0-15 -> M=lane, {v0:K=0, v1:K=1}; lanes 16-31 -> M=lane-16, {v0:K=2, v1:K=3}
// B 4x16 f32 layout: lanes 0-15 -> N=lane, {v0:K=0, v1:K=1}; lanes 16-31 -> N=lane-16, {v0:K=2, v1:K=3}
// C/D: VGPR r -> row M = r + (lane>=16 ? 8 : 0), col N = lane%16
__global__ __launch_bounds__(256) void k_gemm_wmma(const float* __restrict__ X,
                                                   const float* __restrict__ W,
                                                   float* __restrict__ H) {
  const int D = DIM;
  int mt   = blockIdx.x;            // 0..3124  (50000/16)
  int wid  = threadIdx.x >> 5;      // 0..7 -> N tile
  int lane = threadIdx.x & 31;
  int m0 = mt * 16;
  int n0 = wid * 16;
  int lm = lane & 15;
  int hi2 = (lane >= 16) ? 2 : 0;

  const float* arow = X + (size_t)(m0 + lm) * D + hi2;          // A: 2 consecutive K
  const float* bcol = W + (size_t)hi2 * D + n0 + lm;            // B: 2 rows, stride D

  v8f c = {};
#pragma unroll 4
  for (int kk = 0; kk < D; kk += 4) {
    v2f a = *(const v2f*)(arow + kk);                            // K = kk+hi2, kk+hi2+1
    v2f b;
    b.x = bcol[(size_t)kk * D];                                  // K = kk+hi2
    b.y = bcol[(size_t)kk * D + D];                              // K = kk+hi2+1
    c = __builtin_amdgcn_wmma_f32_16x16x4_f32(
        /*neg_a=*/false, a, /*neg_b=*/false, b,
        /*c_mod=*/(short)0, c, /*reuse_a=*/false, /*reuse_b=*/false);
  }

  float* out = H + (size_t)(m0 + ((lane >= 16) ? 8 : 0)) * D + n0 + lm;
#pragma unroll
  for (int r = 0; r < 8; ++r) out[(size_t)r * D] = c[r];
}

// agg[n,d] = self_coef[n] * h[n,d] + b[t,d]   (fused accumulator init: no memset pass)
__global__ void k_agg_init(const float* __restrict__ h, const float* __restrict__ sc,
                           const float* __restrict__ bias, float* __restrict__ agg) {
  int i = blockIdx.x * blockDim.x + threadIdx.x;
  if (i < N_NODES * DIM) {
    int n = i >> 7;
    int d = i & (DIM - 1);
    agg[i] = fmaf(sc[n], h[i], bias[d]);
  }
}

// wave-per-edge scatter: agg[dst] += (a[t,k_e]*norm_e) * h[src]; hardware f32 atomics
__global__ __launch_bounds__(256) void k_edge_scatter(const float* __restrict__ h,
                                                      const float* __restrict__ norm_e,
                                                      const int* __restrict__ src,
                                                      const int* __restrict__ dst,
                                                      const int* __restrict__ ea,
                                                      const float* __restrict__ a_all,
                                                      float* __restrict__ agg, int t) {
  int e    = (blockIdx.x * blockDim.x + threadIdx.x) >> 5;
  int lane = threadIdx.x & 31;
  if (e >= N_EDGES) return;
  int s = src[e], d = dst[e];
  int k = ea[e] - 1;
  float w = a_all[t * K_MAX + k] * norm_e[e];
  const float* hs = h + (size_t)s * DIM;
  float* ad = agg + (size_t)d * DIM;
#pragma unroll
  for (int i = 0; i < DIM / 32; ++i) {
    int c = lane + i * 32;
    unsafeAtomicAdd(&ad[c], w * hs[c]);
  }
}

// wave-per-node: x' = (x + relu(agg)); x' /= max(||x'||_2, 1e-12)
__global__ __launch_bounds__(256) void k_finalize(const float* __restrict__ xin,
                                                  const float* __restrict__ agg,
                                                  float* __restrict__ xout) {
  int n    = (blockIdx.x * blockDim.x + threadIdx.x) >> 5;
  int lane = threadIdx.x & 31;
  if (n >= N_NODES) return;
  const float* xp = xin + (size_t)n * DIM;
  const float* ap = agg + (size_t)n * DIM;
  float v[DIM / 32];
  float ss = 0.0f;
#pragma unroll
  for (int i = 0; i < DIM / 32; ++i) {
    int c = lane + i * 32;
    float o = ap[c];
    o = (o > 0.0f) ? o : 0.0f;
    float xv = xp[c] + o;
    v[i] = xv;
    ss = fmaf(xv, xv, ss);
  }
#pragma unroll
  for (int off = 16; off > 0; off >>= 1) ss += __shfl_xor(ss, off, 32);
  float inv = 1.0f / fmaxf(sqrtf(ss), 1e-12f);
  float* op = xout + (size_t)n * DIM;
#pragma unroll
  for (int i = 0; i < DIM / 32; ++i) op[lane + i * 32] = v[i] * inv;
}

// ---------------- launch ----------------

extern "C" void kernel_launch(void* const* d_in, const int* in_sizes, int n_in,
                              void* d_out, int out_size, void* d_ws, size_t ws_size,
                              hipStream_t stream) {
  const float* x     = (const float*)d_in[0];
  const int*   eidx  = (const int*)d_in[1];    // [2, E]
  const int*   ea    = (const int*)d_in[2];    // [E]
  const float* W     = (const float*)d_in[3];  // [L, D, D]
  const float* bvec  = (const float*)d_in[4];  // [L, D]
  const float* alpha = (const float*)d_in[5];  // [L, K]

  const int* src = eidx;
  const int* dst = eidx + N_EDGES;

  // workspace layout (256B aligned slices)
  char* ws = (char*)d_ws;
  size_t off = 0;
  auto take = [&](size_t bytes) -> char* {
    char* p = ws + off;
    off = (off + bytes + 255) & ~(size_t)255;
    return p;
  };
  float* dinv   = (float*)take((size_t)N_NODES * K_MAX * sizeof(float)); // deg -> rsqrt in place
  float* norm_e = (float*)take((size_t)N_EDGES * sizeof(float));
  float* a_all  = (float*)take((size_t)N_LAYERS * K_MAX * sizeof(float));
  float* sc     = (float*)take((size_t)N_NODES * sizeof(float));
  float* xbuf   = (float*)take((size_t)N_NODES * DIM * sizeof(float));
  float* hbuf   = (float*)take((size_t)N_NODES * DIM * sizeof(float));
  float* agg    = (float*)take((size_t)N_NODES * DIM * sizeof(float));

  const int TB = 256;
  const int ND = N_NODES * DIM;

  // setup
  k_copy<<<(ND + TB - 1) / TB, TB, 0, stream>>>(x, xbuf, ND);
  k_fill1<<<(N_NODES * K_MAX + TB - 1) / TB, TB, 0, stream>>>(dinv, N_NODES * K_MAX);
  k_deg_accum<<<(N_EDGES + TB - 1) / TB, TB, 0, stream>>>(dst, ea, dinv, N_EDGES);
  k_rsqrt_inplace<<<(N_NODES * K_MAX + TB - 1) / TB, TB, 0, stream>>>(dinv, N_NODES * K_MAX);
  k_edge_norm<<<(N_EDGES + TB - 1) / TB, TB, 0, stream>>>(src, dst, ea, dinv, norm_e, N_EDGES);
  k_softmax<<<1, 32, 0, stream>>>(alpha, a_all);

  // layers
  for (int t = 0; t < N_LAYERS; ++t) {
    k_gemm_wmma<<<N_NODES / 16, 256, 0, stream>>>(xbuf, W + (size_t)t * DIM * DIM, hbuf);
    k_selfcoef<<<(N_NODES + TB - 1) / TB, TB, 0, stream>>>(dinv, a_all, sc, t);
    k_agg_init<<<(ND + TB - 1) / TB, TB, 0, stream>>>(hbuf, sc, bvec + (size_t)t * DIM, agg);
    k_edge_scatter<<<(N_EDGES * 32 + TB - 1) / TB, TB, 0, stream>>>(hbuf, norm_e, src, dst, ea,
                                                                    a_all, agg, t);
    float* xout = (t == N_LAYERS - 1) ? (float*)d_out : xbuf;
    k_finalize<<<(N_NODES * 32 + TB - 1) / TB, TB, 0, stream>>>(xbuf, agg, xout);
  }
}